// GATNetwork_8220567405031
// MI455X (gfx1250) — compile-verified
//
#include <hip/hip_runtime.h>
#include <cstddef>

#define HID 256
#define HEADS 8
#define CH 32
#define N_GRAPHS 128
#define OUTF 10
#define NEG_SLOPE 0.2f

typedef float v2f __attribute__((ext_vector_type(2)));
typedef float v8f __attribute__((ext_vector_type(8)));

// ---------------------------------------------------------------------------
// fills
// ---------------------------------------------------------------------------
__global__ void fill_f32(float* __restrict__ p, float v, long n) {
  long i = (long)blockIdx.x * blockDim.x + threadIdx.x;
  if (i < n) p[i] = v;
}
__global__ void fill_i32(int* __restrict__ p, int v, long n) {
  long i = (long)blockIdx.x * blockDim.x + threadIdx.x;
  if (i < n) p[i] = v;
}

// ---------------------------------------------------------------------------
// C[M x 256] = X[M x K] @ W[K x 256], full-precision WMMA (v_wmma_f32_16x16x4_f32)
// One wave computes a 16(M) x 64(N) strip: A fragment reused across 4 WMMAs/step.
// Fragment layouts per CDNA5 ISA 7.12.2:
//   A (16x4 f32):  lane m = lane&15, K = 2*(lane>>4) + vgpr
//   B (4x16 f32):  lane n = lane&15, K = 2*(lane>>4) + vgpr
//   C/D (16x16):   lane n = lane&15, M = vgpr + 8*(lane>>4)
// ---------------------------------------------------------------------------
__global__ void gemm_wmma(const float* __restrict__ X, const float* __restrict__ W,
                          float* __restrict__ C, int M, int K) {
  const int NC = HID;
  int wave = (int)((blockIdx.x * (long)blockDim.x + threadIdx.x) >> 5);
  int lane = threadIdx.x & 31;
  int tm = wave >> 2;                 // 16-row tile index
  int tn = wave & 3;                  // 64-col tile index
  if (tm * 16 >= M) return;

  int half = lane >> 4;               // 0 or 1
  int l15  = lane & 15;
  int row  = tm * 16 + l15;
  if (row >= M) row = M - 1;          // clamp (M is a multiple of 16 here)
  int kofs = half * 2;
  int col  = tn * 64 + l15;

  v8f acc0 = {}; v8f acc1 = {}; v8f acc2 = {}; v8f acc3 = {};
  const float* xp = X + (size_t)row * K;
  for (int k = 0; k < K; k += 4) {
    v2f a; a.x = xp[k + kofs]; a.y = xp[k + kofs + 1];
    const float* w0 = W + (size_t)(k + kofs) * NC + col;
    const float* w1 = w0 + NC;
    v2f b0, b1, b2, b3;
    b0.x = w0[0];  b0.y = w1[0];
    b1.x = w0[16]; b1.y = w1[16];
    b2.x = w0[32]; b2.y = w1[32];
    b3.x = w0[48]; b3.y = w1[48];
    acc0 = __builtin_amdgcn_wmma_f32_16x16x4_f32(false, a, false, b0, 0, acc0, false, false);
    acc1 = __builtin_amdgcn_wmma_f32_16x16x4_f32(false, a, false, b1, 0, acc1, false, false);
    acc2 = __builtin_amdgcn_wmma_f32_16x16x4_f32(false, a, false, b2, 0, acc2, false, false);
    acc3 = __builtin_amdgcn_wmma_f32_16x16x4_f32(false, a, false, b3, 0, acc3, false, false);
  }
#pragma unroll
  for (int v = 0; v < 8; ++v) {
    int rowD = tm * 16 + v + 8 * half;
    if (rowD < M) {
      size_t base = (size_t)rowD * NC + tn * 64 + l15;
      C[base +  0] = acc0[v];
      C[base + 16] = acc1[v];
      C[base + 32] = acc2[v];
      C[base + 48] = acc3[v];
    }
  }
}

// ---------------------------------------------------------------------------
// al_src[n,h] = <h[n,h,:], a_src[h,:]> ; same for a_dst. One thread per (node,head).
// ---------------------------------------------------------------------------
__global__ void attn_logits(const float* __restrict__ h,
                            const float* __restrict__ a_src, const float* __restrict__ a_dst,
                            float* __restrict__ als, float* __restrict__ ald, int n) {
  long t = (long)blockIdx.x * blockDim.x + threadIdx.x;
  if (t >= (long)n * HEADS) return;
  int head = (int)(t & (HEADS - 1));
  long node = t >> 3;
  const float* hp = h + (size_t)node * HID + head * CH;
  const float* as = a_src + head * CH;
  const float* ad = a_dst + head * CH;
  float s = 0.f, d = 0.f;
#pragma unroll
  for (int c = 0; c < CH; ++c) { float v = hp[c]; s += v * as[c]; d += v * ad[c]; }
  als[t] = s; ald[t] = d;
}

__device__ __forceinline__ float lrelu(float v) {
  return v > 0.f ? v : NEG_SLOPE * v;
}

__device__ __forceinline__ void edge_sd(const int* __restrict__ ei, int E, int e,
                                        int& s, int& d) {
  if (e < E) { s = ei[e]; d = ei[E + e]; }   // self-loops appended after real edges
  else       { s = e - E; d = s; }
}

// ---------------------------------------------------------------------------
// CSR build (dst-sorted), done once per call and reused by all 3 layers
// ---------------------------------------------------------------------------
__global__ void deg_count(const int* __restrict__ ei, int E, int n, int* __restrict__ deg) {
  int e = blockIdx.x * blockDim.x + threadIdx.x;
  if (e >= E + n) return;
  int d = (e < E) ? ei[E + e] : (e - E);
  atomicAdd(&deg[d], 1);
}

// exclusive scan of deg[0..n) into offs[0..n], single 1024-thread block
__global__ void scan_excl(const int* __restrict__ deg, int* __restrict__ offs, int n) {
  __shared__ int part[1024];
  int tid = threadIdx.x;
  int chunk = (n + 1023) >> 10;
  int lo = tid * chunk;
  int hi = lo + chunk; if (hi > n) hi = n; if (lo > n) lo = n;
  int sum = 0;
  for (int i = lo; i < hi; ++i) sum += deg[i];
  part[tid] = sum;
  __syncthreads();
  if (tid == 0) {
    int run = 0;
    for (int i = 0; i < 1024; ++i) { int t = part[i]; part[i] = run; run += t; }
  }
  __syncthreads();
  int run = part[tid];
  for (int i = lo; i < hi; ++i) { offs[i] = run; run += deg[i]; }
  if (hi == n) offs[n] = run;   // all tail threads write identical total
}

__global__ void csr_fill(const int* __restrict__ ei, int E, int n,
                         int* __restrict__ cursor, int* __restrict__ csr_src) {
  int e = blockIdx.x * blockDim.x + threadIdx.x;
  if (e >= E + n) return;
  int s, d; edge_sd(ei, E, e, s, d);
  int slot = atomicAdd(&cursor[d], 1);
  csr_src[slot] = s;
}

// ---------------------------------------------------------------------------
// Fused gather-based segment softmax + aggregation + bias.
// One wave32 per (dst node, head); lane == output channel (CH == 32).
//  phase 1: lanes stride incoming edges, wave-reduce max   (shfl_xor butterfly)
//  phase 2: lanes stride incoming edges, wave-reduce sum of exp
//  phase 3: all lanes walk edges together; each lane gathers one channel of
//           h[src] (coalesced 128B/edge) and accumulates in registers.
// No atomics, no zero-init of the output, bias fused into the single store.
// ---------------------------------------------------------------------------
__global__ void gat_gather(const int* __restrict__ offs, const int* __restrict__ csr_src,
                           const float* __restrict__ als, const float* __restrict__ ald,
                           const float* __restrict__ h, const float* __restrict__ bias,
                           float* __restrict__ out, int n) {
  int wave = (int)((blockIdx.x * (long)blockDim.x + threadIdx.x) >> 5);
  int lane = threadIdx.x & 31;
  if (wave >= n * HEADS) return;
  int node = wave >> 3;
  int head = wave & (HEADS - 1);

  int start = offs[node];
  int end   = offs[node + 1];
  float aldv = ald[(size_t)node * HEADS + head];

  // phase 1: segment max
  float mloc = -1e30f;
  for (int j = start + lane; j < end; j += 32)
    mloc = fmaxf(mloc, lrelu(als[(size_t)csr_src[j] * HEADS + head] + aldv));
#pragma unroll
  for (int o = 16; o; o >>= 1) mloc = fmaxf(mloc, __shfl_xor(mloc, o));

  // phase 2: sum of exp
  float sloc = 0.f;
  for (int j = start + lane; j < end; j += 32)
    sloc += __expf(lrelu(als[(size_t)csr_src[j] * HEADS + head] + aldv) - mloc);
#pragma unroll
  for (int o = 16; o; o >>= 1) sloc += __shfl_xor(sloc, o);
  float invd = 1.f / (sloc + 1e-16f);

  // phase 3: weighted gather of h[src, head, lane]
  float acc = 0.f;
  for (int j = start; j < end; ++j) {
    int s = csr_src[j];                         // broadcast load (all lanes same addr)
    float alpha = __expf(lrelu(als[(size_t)s * HEADS + head] + aldv) - mloc) * invd;
    acc += alpha * h[(size_t)s * HID + head * CH + lane];
  }
  out[(size_t)node * HID + head * CH + lane] = acc + bias[head * CH + lane];
}

// ---------------------------------------------------------------------------
// global mean pool + final linear
// ---------------------------------------------------------------------------
__global__ void pool_accum(const float* __restrict__ h, const int* __restrict__ batch,
                           float* __restrict__ sums, float* __restrict__ cnt, int n) {
  long t = (long)blockIdx.x * blockDim.x + threadIdx.x;
  if (t >= (long)n * HID) return;
  long node = t >> 8;                   // HID == 256
  int  c    = (int)(t & (HID - 1));
  int  g    = batch[node];
  atomicAdd(&sums[(size_t)g * HID + c], h[t]);
  if (c == 0) atomicAdd(&cnt[g], 1.0f);
}

__global__ void pool_linear(const float* __restrict__ sums, const float* __restrict__ cnt,
                            const float* __restrict__ Wl, const float* __restrict__ bl,
                            float* __restrict__ out) {
  int t = blockIdx.x * blockDim.x + threadIdx.x;
  if (t >= N_GRAPHS * OUTF) return;
  int g = t / OUTF, o = t - g * OUTF;
  float c = cnt[g]; c = c > 1.f ? c : 1.f;
  float inv = 1.f / c;
  float acc = bl[o];
  const float* sp = sums + (size_t)g * HID;
  for (int k = 0; k < HID; ++k) acc += sp[k] * inv * Wl[k * OUTF + o];
  out[t] = acc;
}

// ---------------------------------------------------------------------------
// orchestration
// ---------------------------------------------------------------------------
static inline int cdiv(long a, long b) { return (int)((a + b - 1) / b); }

extern "C" void kernel_launch(void* const* d_in, const int* in_sizes, int n_in,
                              void* d_out, int out_size, void* d_ws, size_t ws_size,
                              hipStream_t stream) {
  (void)n_in; (void)out_size; (void)ws_size;
  const float* x     = (const float*)d_in[0];
  const int*   ei    = (const int*)d_in[1];
  const int*   batch = (const int*)d_in[2];
  const float* W0    = (const float*)d_in[3];
  const float* as0   = (const float*)d_in[4];
  const float* ad0   = (const float*)d_in[5];
  const float* b0    = (const float*)d_in[6];
  const float* Wh    = (const float*)d_in[7];
  const float* ash   = (const float*)d_in[8];
  const float* adh   = (const float*)d_in[9];
  const float* bh    = (const float*)d_in[10];
  const float* Wl    = (const float*)d_in[11];
  const float* bl    = (const float*)d_in[12];

  const int n    = in_sizes[2];        // 50000 nodes (batch array length)
  const int E    = in_sizes[1] / 2;    // 800000 edges
  const int Etot = E + n;              // + self loops
  const int K0   = in_sizes[0] / n;    // 128 input features

  char* p = (char*)d_ws;
  float* hP      = (float*)p;  p += (size_t)n * HID * sizeof(float);
  float* hQ      = (float*)p;  p += (size_t)n * HID * sizeof(float);
  float* als     = (float*)p;  p += (size_t)n * HEADS * sizeof(float);
  float* ald     = (float*)p;  p += (size_t)n * HEADS * sizeof(float);
  int*   deg     = (int*)p;    p += (size_t)n * sizeof(int);
  int*   offs    = (int*)p;    p += (size_t)(n + 1) * sizeof(int);
  int*   cursor  = (int*)p;    p += (size_t)n * sizeof(int);
  int*   csr_src = (int*)p;    p += (size_t)Etot * sizeof(int);
  float* sums    = (float*)p;  p += (size_t)N_GRAPHS * HID * sizeof(float);
  float* cnt     = (float*)p;  p += (size_t)N_GRAPHS * sizeof(float);

  const int BT = 256;

  // ---- build dst-sorted CSR once (reused by all 3 layers) ----
  fill_i32<<<cdiv(n, BT), BT, 0, stream>>>(deg, 0, n);
  deg_count<<<cdiv(Etot, BT), BT, 0, stream>>>(ei, E, n, deg);
  scan_excl<<<1, 1024, 0, stream>>>(deg, offs, n);
  hipMemcpyAsync(cursor, offs, (size_t)n * sizeof(int), hipMemcpyDeviceToDevice, stream);
  csr_fill<<<cdiv(Etot, BT), BT, 0, stream>>>(ei, E, n, cursor, csr_src);

  // ---- 3 GAT layers ----
  const float* feat = x;
  int K = K0;
  for (int layer = 0; layer < 3; ++layer) {
    const float* W  = layer == 0 ? W0  : Wh  + (size_t)(layer - 1) * HID * HID;
    const float* as = layer == 0 ? as0 : ash + (size_t)(layer - 1) * HEADS * CH;
    const float* ad = layer == 0 ? ad0 : adh + (size_t)(layer - 1) * HEADS * CH;
    const float* b  = layer == 0 ? b0  : bh  + (size_t)(layer - 1) * HID;

    long waves = (long)cdiv(n, 16) * 4;
    gemm_wmma<<<cdiv(waves * 32, BT), BT, 0, stream>>>(feat, W, hP, n, K);
    attn_logits<<<cdiv((long)n * HEADS, BT), BT, 0, stream>>>(hP, as, ad, als, ald, n);

    long gwaves = (long)n * HEADS;
    gat_gather<<<cdiv(gwaves * 32, BT), BT, 0, stream>>>(offs, csr_src, als, ald, hP, b, hQ, n);

    feat = hQ;
    K = HID;
  }

  // ---- global mean pool + final linear ----
  fill_f32<<<cdiv((long)N_GRAPHS * HID, BT), BT, 0, stream>>>(sums, 0.f, (long)N_GRAPHS * HID);
  fill_f32<<<1, BT, 0, stream>>>(cnt, 0.f, N_GRAPHS);
  pool_accum<<<cdiv((long)n * HID, BT), BT, 0, stream>>>(hQ, batch, sums, cnt, n);
  pool_linear<<<cdiv(N_GRAPHS * OUTF, BT), BT, 0, stream>>>(sums, cnt, Wl, bl, (float*)d_out);
}